// bilateral_context_block_80917183857401
// MI455X (gfx1250) — compile-verified
//
#include <hip/hip_runtime.h>
#include <math.h>

// ---------------------------------------------------------------------------
// Bilateral context block, fused for MI455X (gfx1250, wave32, WMMA).
//   B=4, N=16384, K=16, IN_C=32, d=8, D0=16.
// Kernel 1: conv1+bn1+relu -> f[B,N,8] (point-major, in d_ws) for fast gathers.
// Kernel 2: one block per (b, 16-point tile); 256 thr = 16 k x 16 n.
//   phase1: scrambled gathers + conv2..conv5 (VALU, tiny channels)
//   phase2: conv6 as fp32 WMMA 16x16 GEMM over 256 pixels (v_wmma_f32_16x16x4_f32)
//   phase3: softmax over K + weighted-sum + max reductions
//   phase4: conv7 (16x32) + conv8 (32x16) as fp32 WMMA on wave 0, fused BN/acts.
// ---------------------------------------------------------------------------

#define BB   4
#define NN   16384
#define KK   16
#define KN   (KK * NN)      // 262144
#define DD   8
#define EPSF 1e-5f
#define OUT1 (BB * 32 * NN) // 2097152 floats: first output tensor

#define PSTR 257            // [16][256] pixel-plane LDS stride (pad vs bank conflicts)
#define ESTR 17             // [32][16] / [16][16] strides

typedef float v2f __attribute__((ext_vector_type(2)));
typedef float v8f __attribute__((ext_vector_type(8)));

struct P2 {
    const float* f;    // [B,N,8] from kernel 1
    const float* xyz;  // [B,N,3]
    const int*   idx;  // [B,N*K]
    const float *w2, *b2, *w3, *b3, *w4, *b4, *w5, *b5, *w6, *b6, *w7, *b7, *w8, *b8;
    const float *g2, *be2, *m2, *v2;
    const float *g3, *be3, *m3, *v3;
    const float *g4, *be4, *m4, *v4;
    const float *g5, *be5, *m5, *v5;
    const float *g7, *be7, *m7, *v7;
    const float *g8, *be8, *m8, *v8;
    float* out;        // OUT1 floats of out, then B*3*K*N floats of shifted_neigh_xyz
};

// ---------------- kernel 1: conv1 + bn1 + relu -> f[B,N,8] -----------------
__global__ __launch_bounds__(256) void k1_conv1(
    const float* __restrict__ feat, const float* __restrict__ w1,
    const float* __restrict__ b1, const float* __restrict__ g,
    const float* __restrict__ be, const float* __restrict__ m,
    const float* __restrict__ v, float* __restrict__ fout)
{
    int id = blockIdx.x * 256 + threadIdx.x;      // over B*N points
    if (id >= BB * NN) return;
    int b = id >> 14, n = id & (NN - 1);
    const float* fp = feat + (size_t)b * 32 * NN + n;
    float x[32];
#pragma unroll
    for (int i = 0; i < 32; ++i) x[i] = fp[i * NN];
#pragma unroll
    for (int o = 0; o < DD; ++o) {
        float acc = b1[o];
#pragma unroll
        for (int i = 0; i < 32; ++i) acc += w1[o * 32 + i] * x[i];
        float s = g[o] * rsqrtf(v[o] + EPSF);
        float y = (acc - m[o]) * s + be[o];
        fout[(size_t)id * DD + o] = fmaxf(y, 0.f);
    }
}

// ---------------- kernel 2: fused block ------------------------------------
__global__ __launch_bounds__(256) void k2_fused(P2 P)
{
    __shared__ float w2s[48], c2s[3], w3s[72], c3s[8], w4s[72], c4s[8];
    __shared__ float w5s[192], c5s[8], w6s[256], b6s[16];
    __shared__ float w7s[512], c7s[16], w8s[512], c8s[32];
    __shared__ float sc2[3], sc3[8], sc4[8], sc5[8], sc7[16], sc8[32];
    __shared__ float ovs[16 * PSTR];   // overall [ch][px]
    __shared__ float lgs[16 * PSTR];   // conv6 logits [ch][px]
    __shared__ float encs[32 * ESTR];  // enc  [ch][nl]
    __shared__ float enc2s[16 * ESTR]; // conv7 out [ch][nl]

    const int t  = threadIdx.x;
    const int b  = blockIdx.y;
    const int n0 = blockIdx.x * 16;

    // ---- fold BN into weights (scale s = g*rsqrt(v+eps); c = s*(bias-m)+be)
    if (t < 3)  { float s = P.g2[t] * rsqrtf(P.v2[t] + EPSF); sc2[t] = s; c2s[t] = s * (P.b2[t] - P.m2[t]) + P.be2[t]; }
    if (t < 8)  { float s = P.g3[t] * rsqrtf(P.v3[t] + EPSF); sc3[t] = s; c3s[t] = s * (P.b3[t] - P.m3[t]) + P.be3[t]; }
    if (t < 8)  { float s = P.g4[t] * rsqrtf(P.v4[t] + EPSF); sc4[t] = s; c4s[t] = s * (P.b4[t] - P.m4[t]) + P.be4[t]; }
    if (t < 8)  { float s = P.g5[t] * rsqrtf(P.v5[t] + EPSF); sc5[t] = s; c5s[t] = s * (P.b5[t] - P.m5[t]) + P.be5[t]; }
    if (t < 16) { float s = P.g7[t] * rsqrtf(P.v7[t] + EPSF); sc7[t] = s; c7s[t] = s * (P.b7[t] - P.m7[t]) + P.be7[t]; }
    if (t < 32) { float s = P.g8[t] * rsqrtf(P.v8[t] + EPSF); sc8[t] = s; c8s[t] = s * (P.b8[t] - P.m8[t]) + P.be8[t]; }
    if (t < 16) b6s[t] = P.b6[t];
    __syncthreads();
    for (int i = t; i < 48;  i += 256) w2s[i] = P.w2[i] * sc2[i / 16];
    for (int i = t; i < 72;  i += 256) w3s[i] = P.w3[i] * sc3[i / 9];
    for (int i = t; i < 72;  i += 256) w4s[i] = P.w4[i] * sc4[i / 9];
    for (int i = t; i < 192; i += 256) w5s[i] = P.w5[i] * sc5[i / 24];
    for (int i = t; i < 256; i += 256) w6s[i] = P.w6[i];
    for (int i = t; i < 512; i += 256) w7s[i] = P.w7[i] * sc7[i / 32];
    for (int i = t; i < 512; i += 256) w8s[i] = P.w8[i] * sc8[i / 16];
    __syncthreads();

    // ================= phase 1: per-pixel (k, n) work =================
    const int nl = t & 15, k = t >> 4;
    const int n  = n0 + nl;
    const int px = k * 16 + nl;

    const float* fB   = P.f   + (size_t)b * NN * DD;
    const float* xyzB = P.xyz + (size_t)b * NN * 3;
    const int*   idxB = P.idx + (size_t)b * KN;

    float tf[DD], txz[3];
#pragma unroll
    for (int c = 0; c < DD; ++c) tf[c] = fB[n * DD + c];
#pragma unroll
    for (int c = 0; c < 3; ++c) txz[c] = xyzB[n * 3 + c];

    // scrambled gathers (raw reshape of [NK,8] -> [8,K,N])
    float nf[DD];
    {
        const int mbase = k * 2048 + (n >> 3);
        const int chf   = n & 7;
#pragma unroll
        for (int c = 0; c < DD; ++c) {
            int p = idxB[c * 32768 + mbase];
            nf[c] = fB[p * DD + chf];
        }
    }
    // scrambled gathers (raw reshape of [NK,3] -> [3,K,N])
    float nxz[3];
#pragma unroll
    for (int c = 0; c < 3; ++c) {
        unsigned j = (unsigned)(c * KN + k * NN + n);
        unsigned mm = j / 3u;
        int ch = (int)(j - mm * 3u);
        int p  = idxB[mm];
        nxz[c] = xyzB[p * 3 + ch];
    }

    float fi[16];
#pragma unroll
    for (int c = 0; c < DD; ++c) { fi[c] = nf[c] - tf[c]; fi[8 + c] = tf[c]; }

    // conv2+bn2+relu -> shifted_neigh_xyz (output #2)
    float snx[3];
#pragma unroll
    for (int o = 0; o < 3; ++o) {
        float acc = c2s[o];
#pragma unroll
        for (int i = 0; i < 16; ++i) acc += w2s[o * 16 + i] * fi[i];
        float sv = nxz[o] + fmaxf(acc, 0.f);
        snx[o] = sv;
        P.out[OUT1 + (size_t)b * 3 * KN + (size_t)o * KN + k * NN + n] = sv;
    }

    float xi[9];
#pragma unroll
    for (int c = 0; c < 3; ++c) { xi[c] = nxz[c] - txz[c]; xi[3 + c] = snx[c]; xi[6 + c] = txz[c]; }

    float snf[DD], ov[16];
#pragma unroll
    for (int o = 0; o < DD; ++o) {           // conv3 -> shifted_neigh_feat
        float acc = c3s[o];
#pragma unroll
        for (int i = 0; i < 9; ++i) acc += w3s[o * 9 + i] * xi[i];
        snf[o] = nf[o] + fmaxf(acc, 0.f);
    }
#pragma unroll
    for (int o = 0; o < DD; ++o) {           // conv4 -> xyz_enc
        float acc = c4s[o];
#pragma unroll
        for (int i = 0; i < 9; ++i) acc += w4s[o * 9 + i] * xi[i];
        ov[o] = fmaxf(acc, 0.f);
    }
#pragma unroll
    for (int o = 0; o < DD; ++o) {           // conv5 on [snf, fi] -> feat_enc
        float acc = c5s[o];
#pragma unroll
        for (int i = 0; i < 8; ++i)  acc += w5s[o * 24 + i] * snf[i];
#pragma unroll
        for (int i = 0; i < 16; ++i) acc += w5s[o * 24 + 8 + i] * fi[i];
        ov[8 + o] = fmaxf(acc, 0.f);
    }
#pragma unroll
    for (int c = 0; c < 16; ++c) ovs[c * PSTR + px] = ov[c];
    __syncthreads();

    // ================= phase 2: conv6 as fp32 WMMA GEMM =================
    // D[16ch x 256px] = W6[16x16] * overall[16 x 256px], K split into 4x4.
    {
        const int wv = t >> 5, lane = t & 31;
        const int row = lane & 15, hi = lane >> 4;     // A: M=row ; B/D col: N=row
#pragma unroll
        for (int g = 0; g < 2; ++g) {
            const int px0 = wv * 32 + g * 16;
            v8f acc = {};
#pragma unroll
            for (int kc = 0; kc < 4; ++kc) {
                const int kb = kc * 4 + hi * 2;        // A/B K lanes: lo->{0,1}, hi->{2,3}
                v2f a, bb;
                a.x  = w6s[row * 16 + kb];
                a.y  = w6s[row * 16 + kb + 1];
                bb.x = ovs[kb * PSTR + px0 + row];
                bb.y = ovs[(kb + 1) * PSTR + px0 + row];
                acc = __builtin_amdgcn_wmma_f32_16x16x4_f32(
                    false, a, false, bb, (short)0, acc, false, false);
            }
#pragma unroll
            for (int vv = 0; vv < 8; ++vv) {
                int ch = vv + hi * 8;
                lgs[ch * PSTR + px0 + row] = acc[vv] + b6s[ch];
            }
        }
    }
    __syncthreads();

    // ================= phase 3: softmax over K + reductions =================
    {
        const int ch = t >> 4, nn = t & 15;
        float ml = -INFINITY;
#pragma unroll
        for (int kk = 0; kk < KK; ++kk)
            ml = fmaxf(ml, lgs[ch * PSTR + kk * 16 + nn]);
        float se = 0.f, ws = 0.f, om = -INFINITY;
#pragma unroll
        for (int kk = 0; kk < KK; ++kk) {
            float e = expf(lgs[ch * PSTR + kk * 16 + nn] - ml);
            float o = ovs[ch * PSTR + kk * 16 + nn];
            se += e; ws += o * e; om = fmaxf(om, o);
        }
        encs[ch * ESTR + nn]        = om;       // enc[0:16]  = max_k
        encs[(16 + ch) * ESTR + nn] = ws / se;  // enc[16:32] = softmax-weighted sum
    }
    __syncthreads();

    // ================= phase 4: conv7 + conv8 via WMMA (wave 0) =============
    if (t < 32) {
        const int lane = t, row = lane & 15, hi = lane >> 4;
        // conv7: D[16x16] = W7'[16x32] * enc[32x16]  (+c7, relu)
        v8f acc = {};
#pragma unroll
        for (int kc = 0; kc < 8; ++kc) {
            const int kb = kc * 4 + hi * 2;
            v2f a, bb;
            a.x  = w7s[row * 32 + kb];
            a.y  = w7s[row * 32 + kb + 1];
            bb.x = encs[kb * ESTR + row];
            bb.y = encs[(kb + 1) * ESTR + row];
            acc = __builtin_amdgcn_wmma_f32_16x16x4_f32(
                false, a, false, bb, (short)0, acc, false, false);
        }
#pragma unroll
        for (int vv = 0; vv < 8; ++vv) {
            int ch = vv + hi * 8;
            enc2s[ch * ESTR + row] = fmaxf(acc[vv] + c7s[ch], 0.f);
        }
        // conv8: D[32x16] = W8'[32x16] * enc2[16x16] (+c8, leaky 0.01) -> out
#pragma unroll
        for (int rb = 0; rb < 2; ++rb) {
            v8f acc2 = {};
#pragma unroll
            for (int kc = 0; kc < 4; ++kc) {
                const int kb = kc * 4 + hi * 2;
                v2f a, bb;
                a.x  = w8s[(rb * 16 + row) * 16 + kb];
                a.y  = w8s[(rb * 16 + row) * 16 + kb + 1];
                bb.x = enc2s[kb * ESTR + row];
                bb.y = enc2s[(kb + 1) * ESTR + row];
                acc2 = __builtin_amdgcn_wmma_f32_16x16x4_f32(
                    false, a, false, bb, (short)0, acc2, false, false);
            }
#pragma unroll
            for (int vv = 0; vv < 8; ++vv) {
                int co = rb * 16 + vv + hi * 8;
                float z = acc2[vv] + c8s[co];
                z = (z > 0.f) ? z : 0.01f * z;
                P.out[((size_t)b * 32 + co) * NN + n0 + row] = z;
            }
        }
    }
}

// ---------------------------------------------------------------------------
extern "C" void kernel_launch(void* const* d_in, const int* in_sizes, int n_in,
                              void* d_out, int out_size, void* d_ws, size_t ws_size,
                              hipStream_t stream)
{
    (void)in_sizes; (void)n_in; (void)out_size; (void)ws_size;
    // setup_inputs() dict order: feature, input_xyz, input_neigh_idx, then
    // params in insertion order: conv{1..8}_{w,b}, then bn{1,2,3,4,5,7,8}=(g,be,m,v).
    const float* feature = (const float*)d_in[0];
    const float* xyz     = (const float*)d_in[1];
    const int*   nidx    = (const int*)d_in[2];
    int p = 3;
    const float *cw[8], *cb[8];
    for (int i = 0; i < 8; ++i) { cw[i] = (const float*)d_in[p++]; cb[i] = (const float*)d_in[p++]; }
    const float* bn[7][4];
    for (int i = 0; i < 7; ++i)
        for (int j = 0; j < 4; ++j) bn[i][j] = (const float*)d_in[p++];
    // bn rows: 0=bn1 1=bn2 2=bn3 3=bn4 4=bn5 5=bn7 6=bn8

    float* out  = (float*)d_out;
    float* fbuf = (float*)d_ws;   // B*N*8 floats = 2 MB

    k1_conv1<<<(BB * NN + 255) / 256, 256, 0, stream>>>(
        feature, cw[0], cb[0], bn[0][0], bn[0][1], bn[0][2], bn[0][3], fbuf);

    P2 a;
    a.f = fbuf; a.xyz = xyz; a.idx = nidx;
    a.w2 = cw[1]; a.b2 = cb[1]; a.w3 = cw[2]; a.b3 = cb[2];
    a.w4 = cw[3]; a.b4 = cb[3]; a.w5 = cw[4]; a.b5 = cb[4];
    a.w6 = cw[5]; a.b6 = cb[5]; a.w7 = cw[6]; a.b7 = cb[6];
    a.w8 = cw[7]; a.b8 = cb[7];
    a.g2 = bn[1][0]; a.be2 = bn[1][1]; a.m2 = bn[1][2]; a.v2 = bn[1][3];
    a.g3 = bn[2][0]; a.be3 = bn[2][1]; a.m3 = bn[2][2]; a.v3 = bn[2][3];
    a.g4 = bn[3][0]; a.be4 = bn[3][1]; a.m4 = bn[3][2]; a.v4 = bn[3][3];
    a.g5 = bn[4][0]; a.be5 = bn[4][1]; a.m5 = bn[4][2]; a.v5 = bn[4][3];
    a.g7 = bn[5][0]; a.be7 = bn[5][1]; a.m7 = bn[5][2]; a.v7 = bn[5][3];
    a.g8 = bn[6][0]; a.be8 = bn[6][1]; a.m8 = bn[6][2]; a.v8 = bn[6][3];
    a.out = out;

    k2_fused<<<dim3(NN / 16, BB), 256, 0, stream>>>(a);
}